// GCNNet_85882166051005
// MI455X (gfx1250) — compile-verified
//
#include <hip/hip_runtime.h>
#include <cstdint>

typedef __attribute__((ext_vector_type(2))) float v2f;
typedef __attribute__((ext_vector_type(8))) float v8f;

#define N_NODES 100000
#define N_EDGES 1600000
#define IN_DIM 128
#define HID_DIM 64
#define OUT_DIM 40
#define M_TILES (N_NODES / 16)   // 6250, exact

// ---------------------------------------------------------------- utilities
__global__ __launch_bounds__(256) void fill_kernel(float* __restrict__ p, int n, float v) {
    int i = blockIdx.x * blockDim.x + threadIdx.x;
    if (i < n) p[i] = v;
}

__global__ __launch_bounds__(256) void deg_accum_kernel(const long long* __restrict__ dst,
                                                        float* __restrict__ deg, int e) {
    int i = blockIdx.x * blockDim.x + threadIdx.x;
    if (i < e) atomicAdd(&deg[(int)dst[i]], 1.0f);
}

__global__ __launch_bounds__(256) void rsqrt_kernel(float* __restrict__ deg, int n) {
    int i = blockIdx.x * blockDim.x + threadIdx.x;
    if (i < n) deg[i] = rsqrtf(deg[i]);   // deg already includes the +1 self loop
}

// ------------------------------------------------------- GEMM1: h1 = x @ W1
// One wave -> 16 rows x 64 cols. f32 WMMA 16x16x4, K stepped by 4.
__global__ __launch_bounds__(256) void gemm1_wmma(const float* __restrict__ X,
                                                  const float* __restrict__ W,
                                                  float* __restrict__ H) {
    const int wave = blockIdx.x * (blockDim.x >> 5) + (threadIdx.x >> 5);
    if (wave >= M_TILES) return;                      // wave-uniform exit, EXEC stays all-1s
    const int lane = threadIdx.x & 31;
    const int half = lane >> 4;                       // 0: lanes 0-15, 1: lanes 16-31
    const int l    = lane & 15;
    const int m0   = wave * 16;

    const float* arow = X + (size_t)(m0 + l) * IN_DIM;
    v8f c0 = {}, c1 = {}, c2 = {}, c3 = {};

    for (int k = 0; k < IN_DIM; k += 4) {
        const int ka = k + half * 2;                  // this half-wave's K pair
        v2f a; a.x = arow[ka]; a.y = arow[ka + 1];
        const float* wp = W + (size_t)ka * HID_DIM + l;
        v2f b0, b1, b2, b3;
        b0.x = wp[0];  b0.y = wp[HID_DIM + 0];
        b1.x = wp[16]; b1.y = wp[HID_DIM + 16];
        b2.x = wp[32]; b2.y = wp[HID_DIM + 32];
        b3.x = wp[48]; b3.y = wp[HID_DIM + 48];
        c0 = __builtin_amdgcn_wmma_f32_16x16x4_f32(false, a, false, b0, (short)0, c0, false, false);
        c1 = __builtin_amdgcn_wmma_f32_16x16x4_f32(false, a, false, b1, (short)0, c1, false, false);
        c2 = __builtin_amdgcn_wmma_f32_16x16x4_f32(false, a, false, b2, (short)0, c2, false, false);
        c3 = __builtin_amdgcn_wmma_f32_16x16x4_f32(false, a, false, b3, (short)0, c3, false, false);
    }
    // D layout: lanes 0-15 -> M = vgpr, lanes 16-31 -> M = vgpr+8; N = l + n0
    float* crow = H + (size_t)(m0 + half * 8) * HID_DIM + l;
    #pragma unroll
    for (int v = 0; v < 8; ++v) {
        crow[(size_t)v * HID_DIM + 0]  = c0[v];
        crow[(size_t)v * HID_DIM + 16] = c1[v];
        crow[(size_t)v * HID_DIM + 32] = c2[v];
        crow[(size_t)v * HID_DIM + 48] = c3[v];
    }
}

// ---------------------------------------------- GEMM2: h2 = hrelu @ W2 (64->40)
__global__ __launch_bounds__(256) void gemm2_wmma(const float* __restrict__ X,
                                                  const float* __restrict__ W,
                                                  float* __restrict__ H) {
    const int wave = blockIdx.x * (blockDim.x >> 5) + (threadIdx.x >> 5);
    if (wave >= M_TILES) return;
    const int lane = threadIdx.x & 31;
    const int half = lane >> 4;
    const int l    = lane & 15;
    const int m0   = wave * 16;

    const float* arow = X + (size_t)(m0 + l) * HID_DIM;
    v8f c0 = {}, c1 = {}, c2 = {};

    for (int k = 0; k < HID_DIM; k += 4) {
        const int ka = k + half * 2;
        v2f a; a.x = arow[ka]; a.y = arow[ka + 1];
        const float* wp0 = W + (size_t)ka * OUT_DIM;
        const float* wp1 = W + (size_t)(ka + 1) * OUT_DIM;
        v2f b0, b1, b2;
        b0.x = wp0[l];       b0.y = wp1[l];
        b1.x = wp0[16 + l];  b1.y = wp1[16 + l];
        // tile n0=32: only cols 32..39 exist
        b2.x = (32 + l < OUT_DIM) ? wp0[32 + l] : 0.0f;
        b2.y = (32 + l < OUT_DIM) ? wp1[32 + l] : 0.0f;
        c0 = __builtin_amdgcn_wmma_f32_16x16x4_f32(false, a, false, b0, (short)0, c0, false, false);
        c1 = __builtin_amdgcn_wmma_f32_16x16x4_f32(false, a, false, b1, (short)0, c1, false, false);
        c2 = __builtin_amdgcn_wmma_f32_16x16x4_f32(false, a, false, b2, (short)0, c2, false, false);
    }
    float* crow = H + (size_t)(m0 + half * 8) * OUT_DIM + l;
    #pragma unroll
    for (int v = 0; v < 8; ++v) {
        crow[(size_t)v * OUT_DIM + 0]  = c0[v];
        crow[(size_t)v * OUT_DIM + 16] = c1[v];
        if (32 + l < OUT_DIM) crow[(size_t)v * OUT_DIM + 32] = c2[v];
    }
}

// --------------------------------------------------------- edge scatter-add
// Layer 1: 64 consecutive channels per edge (coalesced 256B gather + atomic scatter)
__global__ __launch_bounds__(256) void scatter1_kernel(const long long* __restrict__ src,
                                                       const long long* __restrict__ dst,
                                                       const float* __restrict__ dinv,
                                                       const float* __restrict__ h,
                                                       float* __restrict__ agg) {
    long long flat = (long long)blockIdx.x * blockDim.x + threadIdx.x;
    int e = (int)(flat >> 6);
    int c = (int)(flat & 63);
    if (e >= N_EDGES) return;
    int s = (int)src[e], d = (int)dst[e];
    float nrm = dinv[s] * dinv[d];
    atomicAdd(&agg[(size_t)d * HID_DIM + c], h[(size_t)s * HID_DIM + c] * nrm);
}

__global__ __launch_bounds__(256) void scatter2_kernel(const long long* __restrict__ src,
                                                       const long long* __restrict__ dst,
                                                       const float* __restrict__ dinv,
                                                       const float* __restrict__ h,
                                                       float* __restrict__ agg) {
    long long flat = (long long)blockIdx.x * blockDim.x + threadIdx.x;
    unsigned int uf = (unsigned int)flat;
    int e = (int)(uf / OUT_DIM);
    int c = (int)(uf % OUT_DIM);
    if (flat >= (long long)N_EDGES * OUT_DIM) return;
    int s = (int)src[e], d = (int)dst[e];
    float nrm = dinv[s] * dinv[d];
    atomicAdd(&agg[(size_t)d * OUT_DIM + c], h[(size_t)s * OUT_DIM + c] * nrm);
}

// ----------------------------------------------------------------- epilogues
// hrelu = relu(agg1 + h1*dinv^2 + b1), written in-place into agg1
__global__ __launch_bounds__(256) void finalize1_kernel(float* __restrict__ agg,
                                                        const float* __restrict__ h,
                                                        const float* __restrict__ dinv,
                                                        const float* __restrict__ b) {
    long long flat = (long long)blockIdx.x * blockDim.x + threadIdx.x;
    int i = (int)(flat >> 6);
    int c = (int)(flat & 63);
    if (i >= N_NODES) return;
    float di = dinv[i];
    float v = agg[flat] + h[flat] * (di * di) + b[c];
    agg[flat] = fmaxf(v, 0.0f);
}

// out = agg2 + h2*dinv^2 + b2 (agg2 already accumulated in d_out)
__global__ __launch_bounds__(256) void finalize2_kernel(float* __restrict__ out,
                                                        const float* __restrict__ h,
                                                        const float* __restrict__ dinv,
                                                        const float* __restrict__ b) {
    long long flat = (long long)blockIdx.x * blockDim.x + threadIdx.x;
    unsigned int uf = (unsigned int)flat;
    int i = (int)(uf / OUT_DIM);
    int c = (int)(uf % OUT_DIM);
    if (flat >= (long long)N_NODES * OUT_DIM) return;
    float di = dinv[i];
    out[flat] = out[flat] + h[flat] * (di * di) + b[c];
}

// ------------------------------------------------------------------ launcher
extern "C" void kernel_launch(void* const* d_in, const int* in_sizes, int n_in,
                              void* d_out, int out_size, void* d_ws, size_t ws_size,
                              hipStream_t stream) {
    const float*     x   = (const float*)d_in[0];
    const long long* ei  = (const long long*)d_in[1];   // [2, E] int64 row-major
    const float*     W1  = (const float*)d_in[2];
    const float*     b1  = (const float*)d_in[3];
    const float*     W2  = (const float*)d_in[4];
    const float*     b2  = (const float*)d_in[5];
    float*           out = (float*)d_out;

    const long long* src = ei;
    const long long* dst = ei + N_EDGES;

    // workspace layout (floats): deg/dinv | h1 | agg1(->hrelu) | h2
    float* deg  = (float*)d_ws;                 // N
    float* h1   = deg  + N_NODES;               // N*64
    float* agg1 = h1   + (size_t)N_NODES * HID_DIM;  // N*64
    float* h2   = agg1 + (size_t)N_NODES * HID_DIM;  // N*40

    const int T = 256;
    // init: deg=1 (self loop), agg1=0, out=0
    fill_kernel<<<(N_NODES + T - 1) / T, T, 0, stream>>>(deg, N_NODES, 1.0f);
    fill_kernel<<<(N_NODES * HID_DIM + T - 1) / T, T, 0, stream>>>(agg1, N_NODES * HID_DIM, 0.0f);
    fill_kernel<<<(N_NODES * OUT_DIM + T - 1) / T, T, 0, stream>>>(out, N_NODES * OUT_DIM, 0.0f);

    // degree + rsqrt
    deg_accum_kernel<<<(N_EDGES + T - 1) / T, T, 0, stream>>>(dst, deg, N_EDGES);
    rsqrt_kernel<<<(N_NODES + T - 1) / T, T, 0, stream>>>(deg, N_NODES);

    // layer 1
    const int gemmBlocks = (M_TILES + 7) / 8;   // 8 waves of 32 per block
    gemm1_wmma<<<gemmBlocks, T, 0, stream>>>(x, W1, h1);
    {
        long long work = (long long)N_EDGES * HID_DIM;
        scatter1_kernel<<<(int)((work + T - 1) / T), T, 0, stream>>>(src, dst, deg, h1, agg1);
    }
    {
        long long work = (long long)N_NODES * HID_DIM;
        finalize1_kernel<<<(int)((work + T - 1) / T), T, 0, stream>>>(agg1, h1, deg, b1);
    }

    // layer 2
    gemm2_wmma<<<gemmBlocks, T, 0, stream>>>(agg1, W2, h2);
    {
        long long work = (long long)N_EDGES * OUT_DIM;
        scatter2_kernel<<<(int)((work + T - 1) / T), T, 0, stream>>>(src, dst, deg, h2, out);
    }
    {
        long long work = (long long)N_NODES * OUT_DIM;
        finalize2_kernel<<<(int)((work + T - 1) / T), T, 0, stream>>>(out, h2, deg, b2);
    }
}